// HardnessAwareKernelizedSupCon_71734543777891
// MI455X (gfx1250) — compile-verified
//
#include <hip/hip_runtime.h>
#include <hip/hip_bf16.h>
#include <math.h>

#define B_SZ 2048
#define V_SZ 2
#define D_SZ 256
#define N_SZ 4096
#define TEMP 0.07f
#define EPSV 1e-8f

#define KC 32          // K-chunk staged in LDS
#define LDA 36         // LDS row stride (floats): 144B keeps b128 16B-aligned + bank pad

typedef float v2f __attribute__((ext_vector_type(2)));
typedef float v8f __attribute__((ext_vector_type(8)));
typedef int   v4i __attribute__((ext_vector_type(4)));

#if __has_builtin(__builtin_amdgcn_global_load_async_to_lds_b128)
#define USE_ASYNC_LDS 1
typedef __attribute__((address_space(1))) v4i g_v4i;   // global (printed __device__ by clang)
typedef __attribute__((address_space(3))) v4i l_v4i;   // LDS (printed __shared__)
#endif

// ---- ordered-int key for float atomic min (handles negatives) ----
__device__ __forceinline__ int f2key(float f) {
    int b = __float_as_int(f);
    return (b >= 0) ? b : (b ^ 0x7fffffff);
}
__device__ __forceinline__ float key2f(int k) {
    int b = (k >= 0) ? k : (k ^ 0x7fffffff);
    return __int_as_float(b);
}

// ---------------- kernel 1: init scalars + label min/max ----------------
__global__ __launch_bounds__(256) void k_init(const int* __restrict__ labels,
                                              int* __restrict__ labmm,
                                              int* __restrict__ simminkey,
                                              float* __restrict__ acc) {
    __shared__ int smin[256], smax[256];
    int t = threadIdx.x;
    int lmin = 0x7fffffff, lmax = -0x7fffffff;
    for (int i = t; i < B_SZ; i += 256) {
        int v = labels[i];
        lmin = min(lmin, v);
        lmax = max(lmax, v);
    }
    smin[t] = lmin; smax[t] = lmax;
    __syncthreads();
    for (int s = 128; s > 0; s >>= 1) {
        if (t < s) {
            smin[t] = min(smin[t], smin[t + s]);
            smax[t] = max(smax[t], smax[t + s]);
        }
        __syncthreads();
    }
    if (t == 0) {
        labmm[0] = smin[0];
        labmm[1] = smax[0];
        *simminkey = 0x7fffffff;
        *acc = 0.0f;
    }
}

// ---------------- kernel 2: row L2-normalize (view-major stacking) ----------------
__global__ __launch_bounds__(256) void k_norm(const float* __restrict__ feat,
                                              float* __restrict__ fnorm) {
    __shared__ float red[256];
    int row = blockIdx.x;           // row n of stacked matrix
    int t = threadIdx.x;
    int b = row & (B_SZ - 1);       // n % B
    int v = row >> 11;              // n / B  (B = 2048)
    const float* src = feat + ((size_t)b * V_SZ + v) * D_SZ;
    float x = src[t];
    red[t] = x * x;
    __syncthreads();
    for (int s = 128; s > 0; s >>= 1) {
        if (t < s) red[t] += red[t + s];
        __syncthreads();
    }
    float nrm = sqrtf(red[0]);
    float inv = 1.0f / fmaxf(nrm, 1e-12f);
    fnorm[(size_t)row * D_SZ + t] = x * inv;
}

// ---------------- kernel 3: S = fnorm * fnorm^T via fp32 WMMA + async LDS staging ----
// Workgroup (128 thr = 4 waves) computes a 64x64 tile of S. A-slab (64 rows) and
// B-slab (64 rows, used transposed) are staged in LDS with GLOBAL_LOAD_ASYNC_TO_LDS_B128
// (ASYNCcnt), then each wave computes a 16x64 strip: 4 accumulator tiles, with the
// A fragment reused across 4 WMMAs per K-step.
// Fragment layout for V_WMMA_F32_16X16X4_F32: 16-dim on lane&15; K pair on
// (half, vgpr): lanes 0-15 hold K={k,k+1}, lanes 16-31 hold K={k+2,k+3}.
__global__ __launch_bounds__(128) void k_gemm(const float* __restrict__ f,
                                              float* __restrict__ S) {
    __shared__ float As[64 * LDA];
    __shared__ float Bs[64 * LDA];
    int t = threadIdx.x;
    int wave = t >> 5;
    int lane = t & 31;
    int h = lane >> 4;
    int l = lane & 15;
    int rowA0 = blockIdx.x * 64;
    int rowB0 = blockIdx.y * 64;

    v8f c[4] = {{}, {}, {}, {}};

    for (int k0 = 0; k0 < D_SZ; k0 += KC) {
        // ---- stage 64xKC slabs of A and B into LDS ----
#if defined(USE_ASYNC_LDS)
#pragma unroll
        for (int it = 0; it < 4; ++it) {
            int q = t + 128 * it;       // [0,512): 64 rows x 8 b128-chunks
            int r = q >> 3;
            int cx = (q & 7) * 4;       // float offset within K-chunk
            __builtin_amdgcn_global_load_async_to_lds_b128(
                (g_v4i*)(f + (size_t)(rowA0 + r) * D_SZ + k0 + cx),
                (l_v4i*)&As[r * LDA + cx], 0, 0);
            __builtin_amdgcn_global_load_async_to_lds_b128(
                (g_v4i*)(f + (size_t)(rowB0 + r) * D_SZ + k0 + cx),
                (l_v4i*)&Bs[r * LDA + cx], 0, 0);
        }
#if __has_builtin(__builtin_amdgcn_s_wait_asynccnt)
        __builtin_amdgcn_s_wait_asynccnt(0);
#else
        asm volatile("s_wait_asynccnt 0" ::: "memory");
#endif
#else
#pragma unroll
        for (int it = 0; it < 4; ++it) {
            int q = t + 128 * it;
            int r = q >> 3;
            int cx = (q & 7) * 4;
            float4 va = *(const float4*)(f + (size_t)(rowA0 + r) * D_SZ + k0 + cx);
            float4 vb = *(const float4*)(f + (size_t)(rowB0 + r) * D_SZ + k0 + cx);
            *(float4*)&As[r * LDA + cx] = va;
            *(float4*)&Bs[r * LDA + cx] = vb;
        }
#endif
        __syncthreads();

        // ---- compute: 8 K-steps of 4, A fragment reused across 4 N-tiles ----
#pragma unroll
        for (int kk = 0; kk < KC; kk += 4) {
            v2f a = *(const v2f*)&As[(wave * 16 + l) * LDA + kk + 2 * h];
#pragma unroll
            for (int nt = 0; nt < 4; ++nt) {
                v2f b = *(const v2f*)&Bs[(nt * 16 + l) * LDA + kk + 2 * h];
                c[nt] = __builtin_amdgcn_wmma_f32_16x16x4_f32(
                    /*neg_a=*/false, a, /*neg_b=*/false, b,
                    /*c_mod=*/(short)0, c[nt], /*reuse_a=*/false, /*reuse_b=*/false);
            }
        }
        __syncthreads();
    }

    // ---- store: C VGPR r holds M=r (lanes 0-15) / M=r+8 (lanes 16-31), N=lane&15 ----
#pragma unroll
    for (int nt = 0; nt < 4; ++nt) {
        float* ps = S + (size_t)(rowA0 + wave * 16 + h * 8) * N_SZ + rowB0 + nt * 16 + l;
#pragma unroll
        for (int r = 0; r < 8; ++r)
            ps[(size_t)r * N_SZ] = c[nt][r];
    }
}

// ---------------- kernel 4: per-row max, log-sum-exp, global sim min ----------------
__global__ __launch_bounds__(256) void k_row(const float* __restrict__ S,
                                             float* __restrict__ rowmaxl,
                                             float* __restrict__ unif,
                                             int* __restrict__ simminkey) {
    __shared__ float rmax[256], rmin[256], rsum[256];
    int i = blockIdx.x;
    int t = threadIdx.x;
    const float* row = S + (size_t)i * N_SZ;

    float lmax = -3.4e38f, lmin = 3.4e38f;
    for (int j = t; j < N_SZ; j += 256) {
        float s = row[j];
        lmax = fmaxf(lmax, s);
        lmin = fminf(lmin, s);
    }
    rmax[t] = lmax; rmin[t] = lmin;
    __syncthreads();
    for (int s = 128; s > 0; s >>= 1) {
        if (t < s) {
            rmax[t] = fmaxf(rmax[t], rmax[t + s]);
            rmin[t] = fminf(rmin[t], rmin[t + s]);
        }
        __syncthreads();
    }
    float smax = rmax[0];
    if (t == 0) atomicMin(simminkey, f2key(rmin[0]));

    // sum over j != i of exp((s - smax)/T)
    const float invT = 1.0f / TEMP;
    float se = 0.0f;
    for (int j = t; j < N_SZ; j += 256) {
        if (j == i) continue;
        se += expf((row[j] - smax) * invT);
    }
    rsum[t] = se;
    __syncthreads();
    for (int s = 128; s > 0; s >>= 1) {
        if (t < s) rsum[t] += rsum[t + s];
        __syncthreads();
    }
    if (t == 0) {
        rowmaxl[i] = smax * invT;                 // max of logits
        unif[i] = logf(rsum[0] + EPSV);
    }
}

// ---------------- kernel 5: hardness-weighted masked accumulation ----------------
__global__ __launch_bounds__(256) void k_final(const float* __restrict__ S,
                                               const int* __restrict__ labels,
                                               const float* __restrict__ rowmaxl,
                                               const float* __restrict__ unif,
                                               const int* __restrict__ labmm,
                                               const int* __restrict__ simminkey,
                                               float* __restrict__ acc) {
    __shared__ float rp[256], rw[256];
    int i = blockIdx.x;
    int t = threadIdx.x;
    int li = labels[i & (B_SZ - 1)];
    float rml = rowmaxl[i];
    float un = unif[i];
    float dymax = (float)(labmm[1] - labmm[0]);
    float dzmax = 1.0f - key2f(*simminkey);
    float inv_dz = 1.0f / (dzmax + EPSV);
    float inv_dy = 1.0f / (dymax + EPSV);
    const float invT = 1.0f / TEMP;
    const float* row = S + (size_t)i * N_SZ;

    float ap = 0.0f, aw = 0.0f;
    for (int j = t; j < N_SZ; j += 256) {
        if (j == i) continue;
        int lj = labels[j & (B_SZ - 1)];
        if (lj != li) continue;                    // mask == 0
        float s = row[j];
        float dz = (1.0f - s) * inv_dz;
        float dy = fabsf((float)(li - lj)) * inv_dy;   // 0 here, kept literal
        float u = fminf(fmaxf(dz - dy, -1.0f), 1.0f);
        float m = 1.0f + fmaxf(u, 0.0f);           // LAMBDA_HARD = 1
        float logp = (s * invT - rml) - un;
        ap += m;
        aw += m * logp;
    }
    rp[t] = ap; rw[t] = aw;
    __syncthreads();
    for (int s = 128; s > 0; s >>= 1) {
        if (t < s) { rp[t] += rp[t + s]; rw[t] += rw[t + s]; }
        __syncthreads();
    }
    if (t == 0) {
        float r = rw[0] / fmaxf(rp[0], 1e-8f);
        atomicAdd(acc, r);
    }
}

// ---------------- kernel 6: finalize scalar ----------------
__global__ void k_out(const float* __restrict__ acc, float* __restrict__ out) {
    // loss = -(T/baseT) * mean(mean_log_prob_pos);  T/baseT == 1
    out[0] = -(*acc) / (float)N_SZ;
}

extern "C" void kernel_launch(void* const* d_in, const int* in_sizes, int n_in,
                              void* d_out, int out_size, void* d_ws, size_t ws_size,
                              hipStream_t stream) {
    const float* features = (const float*)d_in[0];
    const int* labels = (const int*)d_in[1];
    float* out = (float*)d_out;

    // workspace layout
    float* fnorm = (float*)d_ws;                          // N*D
    float* S = fnorm + (size_t)N_SZ * D_SZ;               // N*N
    float* rowmaxl = S + (size_t)N_SZ * N_SZ;             // N
    float* unif = rowmaxl + N_SZ;                         // N
    int* labmm = (int*)(unif + N_SZ);                     // 2
    int* simminkey = labmm + 2;                           // 1
    float* acc = (float*)(simminkey + 1);                 // 1

    k_init<<<1, 256, 0, stream>>>(labels, labmm, simminkey, acc);
    k_norm<<<N_SZ, 256, 0, stream>>>(features, fnorm);
    k_gemm<<<dim3(N_SZ / 64, N_SZ / 64), 128, 0, stream>>>(fnorm, S);
    k_row<<<N_SZ, 256, 0, stream>>>(S, rowmaxl, unif, simminkey);
    k_final<<<N_SZ, 256, 0, stream>>>(S, labels, rowmaxl, unif, labmm, simminkey, acc);
    k_out<<<1, 1, 0, stream>>>(acc, out);
}